// ProbAttention_42193758716146
// MI455X (gfx1250) — compile-verified
//
#include <hip/hip_runtime.h>
#include <hip/hip_bf16.h>

typedef float v2f __attribute__((ext_vector_type(2)));
typedef float v8f __attribute__((ext_vector_type(8)));

#define NEGINF (-1.0e30f)
#define MAXUT 8   // supports u up to 128 selected queries

// ---------------------------------------------------------------------------
// Kernel 1: M[b,h,l] = max_s(Q[b,l,h,:].K[b,idx[l,s],h,:]) - mean_s(...)
// One wave (32 lanes) per query; lane holds 2 dims of Q; wave-reduce dots.
// ---------------------------------------------------------------------------
__global__ void probattn_mscore_kernel(const float* __restrict__ Q,
                                       const float* __restrict__ K,
                                       const int* __restrict__ idx,
                                       float* __restrict__ M,
                                       int B, int H, int L, int S) {
    int wave = threadIdx.x >> 5;
    int lane = threadIdx.x & 31;
    int qid  = blockIdx.x * (blockDim.x >> 5) + wave;   // linear over (b,h,l)
    int total = B * H * L;
    if (qid >= total) return;
    int l = qid % L;
    int h = (qid / L) % H;
    int b = qid / (H * L);

    const float2 qv = *reinterpret_cast<const float2*>(
        Q + ((((size_t)b * L + l) * H + h) * 64) + 2 * lane);

    float mx = NEGINF;
    float sm = 0.0f;
    for (int s = 0; s < S; ++s) {
        int ki = idx[(size_t)l * S + s];
        const float2 kv = *reinterpret_cast<const float2*>(
            K + ((((size_t)b * L + ki) * H + h) * 64) + 2 * lane);
        float d = qv.x * kv.x + qv.y * kv.y;
        // full-wave (32) sum reduction
        for (int off = 16; off > 0; off >>= 1)
            d += __shfl_xor(d, off, 32);
        mx = fmaxf(mx, d);
        sm += d;
    }
    if (lane == 0)
        M[qid] = mx - sm / (float)S;
}

// ---------------------------------------------------------------------------
// Kernel 2: per (b,h) iterative argmax top-u over M[bh,0..L-1]; destructive.
// Tie-break: smallest index (matches stable top_k ordering semantics; order
// of M_top does not affect the final scatter since indices are unique).
// ---------------------------------------------------------------------------
__global__ void probattn_topk_kernel(float* __restrict__ M,
                                     int* __restrict__ Mtop,
                                     int L, int u) {
    __shared__ float sval[256];
    __shared__ int   sidx[256];
    float* Mb = M + (size_t)blockIdx.x * L;
    int tid = threadIdx.x;

    for (int it = 0; it < u; ++it) {
        float bv = -3.0e38f;
        int   bi = 0x7fffffff;
        for (int i = tid; i < L; i += 256) {
            float v = Mb[i];
            if (v > bv || (v == bv && i < bi)) { bv = v; bi = i; }
        }
        sval[tid] = bv; sidx[tid] = bi;
        __syncthreads();
        for (int s = 128; s > 0; s >>= 1) {
            if (tid < s) {
                float ov = sval[tid + s]; int oi = sidx[tid + s];
                if (ov > sval[tid] || (ov == sval[tid] && oi < sidx[tid])) {
                    sval[tid] = ov; sidx[tid] = oi;
                }
            }
            __syncthreads();
        }
        if (tid == 0) {
            Mtop[(size_t)blockIdx.x * u + it] = sidx[0];
            Mb[sidx[0]] = -3.0e38f;   // remove winner
        }
        __syncthreads();
    }
}

// ---------------------------------------------------------------------------
// Kernel 3: baseline context = cumsum(V) over L, written to out[b,l,h,d].
// Block per (b,h): 256 threads = 4 L-chunks x 64 dims, two-phase scan,
// fully coalesced 256B accesses.
// ---------------------------------------------------------------------------
__global__ void probattn_cumsum_kernel(const float* __restrict__ V,
                                       float* __restrict__ out,
                                       int B, int H, int L) {
    int bh = blockIdx.x;
    int b = bh / H, h = bh % H;
    int d = threadIdx.x & 63;
    int c = threadIdx.x >> 6;          // chunk 0..3
    int chunkL = L >> 2;
    __shared__ float csum[4][64];

    const size_t stride = (size_t)H * 64;
    const float* vbase = V + ((size_t)b * L * H + h) * 64 + d;
    float*       obase = out + ((size_t)b * L * H + h) * 64 + d;

    float acc = 0.0f;
    int l0 = c * chunkL, l1 = l0 + chunkL;
    for (int l = l0; l < l1; ++l)
        acc += vbase[(size_t)l * stride];
    csum[c][d] = acc;
    __syncthreads();

    float prefix = 0.0f;
    for (int cc = 0; cc < c; ++cc) prefix += csum[cc][d];

    acc = prefix;
    for (int l = l0; l < l1; ++l) {
        acc += vbase[(size_t)l * stride];
        obase[(size_t)l * stride] = acc;
    }
}

// ---------------------------------------------------------------------------
// Kernel 4: sparse causal attention for the top-u queries, flash-style with
// V_WMMA_F32_16X16X4_F32. One workgroup of UT waves per (b,h); wave w owns
// query rows [16w, 16w+16). Streams 16-key blocks through LDS; online
// softmax kept in registers via width-16 shuffles.
// ---------------------------------------------------------------------------
__global__ void probattn_flash_kernel(const float* __restrict__ Q,
                                      const float* __restrict__ K,
                                      const float* __restrict__ V,
                                      const int* __restrict__ Mtop,
                                      float* __restrict__ out,
                                      int B, int H, int L, int u) {
    __shared__ float Klds[16][68];           // 16 keys x 64 dims, padded
    __shared__ float Vlds[16][68];
    __shared__ float Plds[MAXUT][16][17];    // per-wave P transpose staging
    __shared__ int   qidx_sh[MAXUT * 16];
    __shared__ int   qmax_sh;

    const int bh = blockIdx.x;
    const int b = bh / H, h = bh % H;
    const int tid = threadIdx.x;
    const int wave = tid >> 5;
    const int lane = tid & 31;
    const int lhalf = lane & 15;             // column / M index within half
    const int ghalf = lane >> 4;             // 0 or 1
    const int UT = blockDim.x >> 5;
    const float scale = 0.125f;              // 1/sqrt(64)

    // ---- load selected query indices (pad rows -> index 0) ----
    for (int i = tid; i < UT * 16; i += blockDim.x)
        qidx_sh[i] = (i < u) ? Mtop[(size_t)bh * u + i] : 0;
    __syncthreads();
    if (tid == 0) {
        int qm = 0;
        for (int i = 0; i < u; ++i) qm = max(qm, qidx_sh[i]);
        qmax_sh = qm;
    }
    __syncthreads();

    // ---- per-wave register setup ----
    const int qrow = qidx_sh[wave * 16 + lhalf];    // query for A-row lhalf
    const float* qptr = Q + (((size_t)b * L + qrow) * H + h) * 64;
    v2f afrag[16];
    #pragma unroll
    for (int s = 0; s < 16; ++s) {
        int dd = 4 * s + 2 * ghalf;
        float2 t = *reinterpret_cast<const float2*>(qptr + dd);
        afrag[s].x = t.x; afrag[s].y = t.y;
    }

    int qi[8];
    #pragma unroll
    for (int j = 0; j < 8; ++j)
        qi[j] = qidx_sh[wave * 16 + j + 8 * ghalf];

    // wave-local max query index (for causal block skipping)
    int wqmax = 0;
    #pragma unroll
    for (int j = 0; j < 8; ++j) wqmax = max(wqmax, qi[j]);
    for (int off = 16; off > 0; off >>= 1)
        wqmax = max(wqmax, __shfl_xor(wqmax, off, 32));

    float mrow[8], lrow[8];
    #pragma unroll
    for (int j = 0; j < 8; ++j) { mrow[j] = NEGINF; lrow[j] = 0.0f; }
    v8f Ofrag[4];
    #pragma unroll
    for (int t = 0; t < 4; ++t)
        Ofrag[t] = (v8f){0.f,0.f,0.f,0.f,0.f,0.f,0.f,0.f};

    const int nkb = (qmax_sh + 16) >> 4;     // block-uniform loop bound

    for (int kb = 0; kb < nkb; ++kb) {
        // ---- cooperative load of 16-key K/V tiles (float4 granularity) ----
        for (int i = tid; i < 16 * 16; i += blockDim.x) {
            int r = i >> 4, c4 = i & 15;
            int key = kb * 16 + r;
            size_t goff = (((size_t)b * L + key) * H + h) * 64 + c4 * 4;
            float4 kd = *reinterpret_cast<const float4*>(K + goff);
            float4 vd = *reinterpret_cast<const float4*>(V + goff);
            *reinterpret_cast<float4*>(&Klds[r][c4 * 4]) = kd;
            *reinterpret_cast<float4*>(&Vlds[r][c4 * 4]) = vd;
        }
        __syncthreads();

        if (kb * 16 <= wqmax) {   // wave-uniform predicate: EXEC all-ones inside
            // ---- S = Q . K^T  (16 K-steps of K=4 over D=64) ----
            v8f S = (v8f){0.f,0.f,0.f,0.f,0.f,0.f,0.f,0.f};
            #pragma unroll
            for (int s = 0; s < 16; ++s) {
                v2f bf = *reinterpret_cast<const v2f*>(&Klds[lhalf][4 * s + 2 * ghalf]);
                S = __builtin_amdgcn_wmma_f32_16x16x4_f32(
                        false, afrag[s], false, bf, (short)0, S, false, false);
            }

            // ---- scale + causal mask ----
            const int colkey = kb * 16 + lhalf;
            float sv[8];
            #pragma unroll
            for (int j = 0; j < 8; ++j) {
                float x = S[j] * scale;
                sv[j] = (colkey > qi[j]) ? NEGINF : x;
            }

            // ---- online softmax (row stats via width-16 shuffles) ----
            float rmax[8], p[8], rsum[8], fac[8];
            #pragma unroll
            for (int j = 0; j < 8; ++j) {
                float r = sv[j];
                for (int off = 8; off > 0; off >>= 1)
                    r = fmaxf(r, __shfl_xor(r, off, 16));
                rmax[j] = r;
            }
            #pragma unroll
            for (int j = 0; j < 8; ++j) {
                float mn = fmaxf(mrow[j], rmax[j]);
                fac[j] = __expf(mrow[j] - mn);
                mrow[j] = mn;
                p[j] = __expf(sv[j] - mn);
            }
            #pragma unroll
            for (int j = 0; j < 8; ++j) {
                float r = p[j];
                for (int off = 8; off > 0; off >>= 1)
                    r += __shfl_xor(r, off, 16);
                rsum[j] = r;
                lrow[j] = lrow[j] * fac[j] + rsum[j];
            }
            #pragma unroll
            for (int t = 0; t < 4; ++t)
                #pragma unroll
                for (int j = 0; j < 8; ++j)
                    Ofrag[t][j] *= fac[j];

            // ---- stage P through LDS to get A-matrix layout ----
            #pragma unroll
            for (int j = 0; j < 8; ++j)
                Plds[wave][j + 8 * ghalf][lhalf] = p[j];
            v2f pf[4];
            #pragma unroll
            for (int s = 0; s < 4; ++s) {
                int k0 = 4 * s + 2 * ghalf;
                pf[s].x = Plds[wave][lhalf][k0];
                pf[s].y = Plds[wave][lhalf][k0 + 1];
            }

            // ---- O += P . V  (4 dim-tiles x 4 K-steps) ----
            #pragma unroll
            for (int t = 0; t < 4; ++t) {
                v8f acc = Ofrag[t];
                #pragma unroll
                for (int s = 0; s < 4; ++s) {
                    int k0 = 4 * s + 2 * ghalf;
                    v2f bf;
                    bf.x = Vlds[k0][16 * t + lhalf];
                    bf.y = Vlds[k0 + 1][16 * t + lhalf];
                    acc = __builtin_amdgcn_wmma_f32_16x16x4_f32(
                            false, pf[s], false, bf, (short)0, acc, false, false);
                }
                Ofrag[t] = acc;
            }
        }
        __syncthreads();
    }

    // ---- normalize and scatter into the cumsum baseline ----
    #pragma unroll
    for (int j = 0; j < 8; ++j) {
        int row = wave * 16 + j + 8 * ghalf;
        if (row < u) {
            float invl = 1.0f / lrow[j];
            size_t obase = (((size_t)b * L + qi[j]) * H + h) * 64;
            #pragma unroll
            for (int t = 0; t < 4; ++t)
                out[obase + 16 * t + lhalf] = Ofrag[t][j] * invl;
        }
    }
}

// ---------------------------------------------------------------------------
extern "C" void kernel_launch(void* const* d_in, const int* in_sizes, int n_in,
                              void* d_out, int out_size, void* d_ws, size_t ws_size,
                              hipStream_t stream) {
    const float* Q  = (const float*)d_in[0];
    const float* K  = (const float*)d_in[1];
    const float* V  = (const float*)d_in[2];
    const int*  idx = (const int*)d_in[3];

    const int B = 4, H = 8, L = 2048;
    int S = in_sizes[3] / L;         // sample_k (= 40)
    if (S < 1) S = 1;
    int u = S;                        // u uses the same formula as sample_k
    if (u > MAXUT * 16) u = MAXUT * 16;

    float* Mws  = (float*)d_ws;                       // B*H*L floats
    int*   Mtop = (int*)(Mws + (size_t)B * H * L);    // B*H*u ints

    // 1) sampled max-minus-mean scores
    int totalWaves = B * H * L;
    probattn_mscore_kernel<<<(totalWaves + 7) / 8, 256, 0, stream>>>(
        Q, K, idx, Mws, B, H, L, S);

    // 2) top-u query selection per (b,h)
    probattn_topk_kernel<<<B * H, 256, 0, stream>>>(Mws, Mtop, L, u);

    // 3) baseline: cumulative sum of V along L
    probattn_cumsum_kernel<<<B * H, 256, 0, stream>>>(V, (float*)d_out, B, H, L);

    // 4) WMMA flash attention for selected queries, scattered over baseline
    int UT = (u + 15) >> 4;
    probattn_flash_kernel<<<B * H, 32 * UT, 0, stream>>>(
        Q, K, V, Mtop, (float*)d_out, B, H, L, u);
}